// FNO2d_small_64080912056959
// MI455X (gfx1250) — compile-verified
//
#include <hip/hip_runtime.h>

// FNO2d forward for MI455X (gfx1250, wave32, WMMA).
// All GEMM-like stages use V_WMMA_F32_16X16X4_F32 (fp32 in/out, 8-VGPR C tile).
// LDS staging uses GLOBAL_LOAD_ASYNC_TO_LDS_B128 (ASYNCcnt) when available.

typedef __attribute__((ext_vector_type(2))) float v2f;
typedef __attribute__((ext_vector_type(8))) float v8f;
typedef __attribute__((ext_vector_type(4))) int   v4i;

#define B_    8
#define H_    512
#define W_    512
#define C_    16
#define MODES_ 8
#define HID_  64
#define NPIX  (B_*H_*W_)   /* 2097152 */
#define ROWS  (B_*H_)      /* 4096 */

// workspace layout (float offsets)
#define WS_H   0UL          /* h buffer [B][H][W][16]  : 33,554,432 f */
#define WS_G   33554432UL   /* G [B][H][16][16]        :  1,048,576 f */
#define WS_Z   34603008UL   /* Z [B][H][16][16]        :  1,048,576 f */
#define WS_TT  35651584UL   /* fwd twiddle [512][16]   :      8,192 f */
#define WS_U   35659776UL   /* inv twiddle [512][16]   :      8,192 f */
/* total ~142.7 MB */

#define TWOPI_512 0.012271846303085130f  /* 2*pi/512 */

static __device__ __forceinline__ v8f wmma4(v2f a, v2f b, v8f c) {
  // D = A(16x4,f32) * B(4x16,f32) + C(16x16,f32)
  return __builtin_amdgcn_wmma_f32_16x16x4_f32(false, a, false, b, (short)0, c,
                                               false, false);
}

// --- async global -> LDS staging (CDNA5 ASYNCcnt path), with safe fallback ---
#if defined(__AMDGCN__) && __has_builtin(__builtin_amdgcn_global_load_async_to_lds_b128) && \
    __has_builtin(__builtin_amdgcn_s_wait_asynccnt)
#define HAVE_ASYNC_LDS 1
#endif

static __device__ __forceinline__ void lds_copy_b128(const float* g, float* l) {
#if HAVE_ASYNC_LDS
  __builtin_amdgcn_global_load_async_to_lds_b128(
      (__attribute__((address_space(1))) v4i*)g,
      (__attribute__((address_space(3))) v4i*)l, 0, 0);
#else
  *(float4*)l = *(const float4*)g;
#endif
}
static __device__ __forceinline__ void lds_copy_wait() {
#if HAVE_ASYNC_LDS
  __builtin_amdgcn_s_wait_asynccnt(0);
#endif
}

// ---------------------------------------------------------------------------
// Twiddle tables.
// TT[w*16+r]: r<8 -> cos(2pi r w/512)/(H*W), r>=8 -> -sin(2pi (r-8) w/512)/(H*W)
// UT[w*16+r]: r<8 -> c_ky*cos(.), r>=8 -> -c_ky*sin(.), c_0=1, c_ky=2 otherwise
// ---------------------------------------------------------------------------
__global__ void fno_init(float* __restrict__ tt, float* __restrict__ ut) {
  int i = blockIdx.x * blockDim.x + threadIdx.x;    // 16384 threads
  if (i >= 2 * H_ * C_) return;
  int tab = i >> 13;
  int j = i & 8191;
  int w = j >> 4, r = j & 15, ky = r & 7;
  float s, c;
  __sincosf(TWOPI_512 * (float)((ky * w) & 511), &s, &c);
  if (tab == 0) {
    const float sc = 1.0f / ((float)H_ * (float)W_);
    tt[j] = (r < 8 ? c : -s) * sc;
  } else {
    float cf = (ky == 0) ? 1.0f : 2.0f;
    ut[j] = (r < 8) ? cf * c : -cf * s;
  }
}

// ---------------------------------------------------------------------------
// fc0: lift 1 -> 16 channels, channels-last h[p][16]
// ---------------------------------------------------------------------------
__global__ void fno_fc0(const float* __restrict__ x, const float* __restrict__ w0,
                        const float* __restrict__ b0, float* __restrict__ hb) {
  int p = blockIdx.x * blockDim.x + threadIdx.x;    // exact grid
  float xv = x[p];
  float4* o = (float4*)(hb + (size_t)p * C_);
  const float4* wv = (const float4*)w0;
  const float4* bv = (const float4*)b0;
#pragma unroll
  for (int q = 0; q < 4; ++q) {
    float4 w4 = wv[q], b4 = bv[q];
    o[q] = make_float4(fmaf(xv, w4.x, b4.x), fmaf(xv, w4.y, b4.y),
                       fmaf(xv, w4.z, b4.z), fmaf(xv, w4.w, b4.w));
  }
}

// ---------------------------------------------------------------------------
// Stage A: per (b,h) row, G[ky2][ci] = sum_w T[ky2][w] * h[w][ci]
// GEMM M=16,K=512,N=16. One wave per row, 4 rows per 128-thread block.
// A frags come from the LDS twiddle table, B frags stream from global h.
// ---------------------------------------------------------------------------
__global__ void fno_stageA(const float* __restrict__ hb, const float* __restrict__ tt,
                           float* __restrict__ G) {
  __shared__ float lt[H_ * C_];   // 32KB: twiddles, [w][ky2]
  int tid = threadIdx.x;
  for (int i = tid; i < (H_ * C_) / 4; i += 128)
    lds_copy_b128(tt + i * 4, lt + i * 4);
  lds_copy_wait();
  __syncthreads();

  int wave = tid >> 5, lane = tid & 31, lm = lane & 15, hi = lane >> 4;
  int row = blockIdx.x * 4 + wave;                  // (b*512+h)
  const float* rb = hb + (size_t)row * (W_ * C_);

  v8f acc = {};
  for (int k0 = 0; k0 < W_; k0 += 64) {
    __builtin_prefetch(rb + (size_t)(k0 + 64) * C_, 0, 1);  // global_prefetch_b8
#pragma unroll
    for (int kk = 0; kk < 64; kk += 4) {
      int k = k0 + kk;
      v2f a, b;
      a.x = lt[(k + hi * 2) * C_ + lm];       // A: M=lane%16(ky2), K split per half
      a.y = lt[(k + hi * 2 + 1) * C_ + lm];
      b.x = rb[(k + hi) * C_ + lm];           // B: rows k..k+3 of [w][ci]
      b.y = rb[(k + 2 + hi) * C_ + lm];
      acc = wmma4(a, b, acc);
    }
  }
  float* gb = G + (size_t)row * 256;          // G[b][h][ky2][ci]
#pragma unroll
  for (int v = 0; v < 8; ++v) gb[(v + 8 * hi) * 16 + lm] = acc[v];
}

// ---------------------------------------------------------------------------
// Column DFT + complex mode mixing + inverse column DFT, one block per batch.
//  X[kx2][ky*16+ci] = sum_h twiddle * G   (GEMM M=16,K=1024,N=128, X in LDS)
//  Y = X x Wspec (complex, per-mode 16x16 channel mix), packed into BB in LDS
//  Z[h][ky2][co]  = sum_kx twiddle * BB   (GEMM M=512,K=16,N=256)
// ---------------------------------------------------------------------------
__global__ void fno_specmix(const float* __restrict__ G, const float* __restrict__ swr,
                            const float* __restrict__ swi, float* __restrict__ Z) {
  extern __shared__ float sm[];
  float* Atab = sm;             // [1024][16]  phase 1      (64KB)
  float* Xs   = sm + 16384;     // [16][128]   phases 1-2   (8KB)
  float* BB   = sm + 18432;     // [16][256]   phases 2-3   (16KB)
  float* A2   = sm;             // [512][17]   phases 2-3 (reuses Atab)

  int tid = threadIdx.x;
  int b = blockIdx.x;

  // forward column-DFT twiddles: A[kx2][2h+part]
  for (int i = tid; i < 16384; i += 256) {
    int kk = i >> 4, m = i & 15;
    int h = kk >> 1, part = kk & 1, kx = m & 7;
    float s, c;
    __sincosf(TWOPI_512 * (float)((kx * h) & 511), &s, &c);
    Atab[i] = (m < 8) ? (part ? s : c) : (part ? c : -s);
  }
  __syncthreads();

  int wave = tid >> 5, lane = tid & 31, lm = lane & 15, hi = lane >> 4;
  const float* Gb = G + (size_t)b * (H_ * 256);
  {
    v8f acc = {};
    int col = wave * 16 + lm;   // N column (ky*16+ci), 8 waves cover N=128
    for (int kk = 0; kk < 1024; kk += 4) {
      v2f a, bb;
      a.x = Atab[(kk + hi * 2) * 16 + lm];
      a.y = Atab[(kk + hi * 2 + 1) * 16 + lm];
      int r0 = kk + hi, r1 = kk + 2 + hi;     // K index = 2h+part
      bb.x = Gb[(r0 >> 1) * 256 + (r0 & 1) * 128 + col];
      bb.y = Gb[(r1 >> 1) * 256 + (r1 & 1) * 128 + col];
      acc = wmma4(a, bb, acc);
    }
#pragma unroll
    for (int v = 0; v < 8; ++v) Xs[(v + 8 * hi) * 128 + col] = acc[v];
  }
  __syncthreads();

  // complex 16x16 channel mix per (kx,ky); pack Zr/Zi combinations into BB
  for (int it = 0; it < 4; ++it) {
    int item = tid + 256 * it;                 // 1024 = 8kx * 8ky * 16co
    int co = item & 15, ky = (item >> 4) & 7, kx = item >> 7;
    float yr = 0.f, yi = 0.f;
    for (int ci = 0; ci < 16; ++ci) {
      float xr = Xs[kx * 128 + ky * 16 + ci];
      float xi = Xs[(kx + 8) * 128 + ky * 16 + ci];
      int widx = ((ci * 16 + co) * 8 + kx) * 8 + ky;
      float wr = swr[widx], wim = swi[widx];
      yr = fmaf(xr, wr, yr); yr = fmaf(-xi, wim, yr);
      yi = fmaf(xr, wim, yi); yi = fmaf(xi, wr, yi);
    }
    int n = ky * 16 + co;
    BB[kx * 256 + n]             = yr;   // Zr cols: cos*Yr
    BB[(kx + 8) * 256 + n]       = -yi;  //          sin*(-Yi)
    BB[kx * 256 + 128 + n]       = yi;   // Zi cols: cos*Yi
    BB[(kx + 8) * 256 + 128 + n] = yr;   //          sin*Yr
  }
  // inverse column-DFT twiddles: A2[h][kx2], stride 17 to dodge bank conflicts
  for (int i = tid; i < 8192; i += 256) {
    int h = i >> 4, m = i & 15, kx = m & 7;
    float s, c;
    __sincosf(TWOPI_512 * (float)((kx * h) & 511), &s, &c);
    A2[h * 17 + m] = (m < 8) ? c : s;
  }
  __syncthreads();

  // inverse column DFT: M=512(h), K=16, N=256 -> Z[b][h][ky2][co]
  float* Zb = Z + (size_t)b * (H_ * 256);
  for (int i = 0; i < 64; ++i) {
    int t2 = wave + i * 8;
    int mt = t2 >> 4, nt = t2 & 15;            // nt == ky2 of output
    v8f acc = {};
#pragma unroll
    for (int k = 0; k < 16; k += 4) {
      v2f a, bb;
      a.x = A2[(mt * 16 + lm) * 17 + k + hi * 2];
      a.y = A2[(mt * 16 + lm) * 17 + k + hi * 2 + 1];
      bb.x = BB[(k + hi) * 256 + nt * 16 + lm];
      bb.y = BB[(k + 2 + hi) * 256 + nt * 16 + lm];
      acc = wmma4(a, bb, acc);
    }
#pragma unroll
    for (int v = 0; v < 8; ++v) {
      int h = mt * 16 + v + 8 * hi;
      Zb[(h * 16 + nt) * 16 + lm] = acc[v];
    }
  }
}

// ---------------------------------------------------------------------------
// Inverse row DFT + 1x1 conv + bias, fused, in place.
// h'[w][co] = sum_ky2 U[w][ky2]*Z[ky2][co] + sum_ci h[w][ci]*Wt[ci][co] + b[co]
// Both GEMMs (K=16,N=16,M=512) accumulate into the same WMMA tile.
// Row is staged into LDS (async global->LDS) so the in-place write is safe.
// ---------------------------------------------------------------------------
__global__ void fno_inv2_local(float* __restrict__ hb, const float* __restrict__ Z,
                               const float* __restrict__ ut, const float* __restrict__ cw,
                               const float* __restrict__ cb) {
  extern __shared__ float sm[];
  float* hr = sm;          // [512][17]
  float* Ut = sm + 8704;   // [512][17]
  int tid = threadIdx.x;
  int row = blockIdx.x;
  float* rb = hb + (size_t)row * (W_ * C_);
  // async-stage the row and the U table, 16B chunks into padded LDS rows
  for (int i = tid; i < (W_ * C_) / 4; i += 128) {
    int w = i >> 2, c = (i & 3) * 4;           // chunk fits within a padded row
    lds_copy_b128(rb + i * 4, hr + w * 17 + c);
    lds_copy_b128(ut + i * 4, Ut + w * 17 + c);
  }
  int lane = tid & 31, lm = lane & 15, hi = lane >> 4, wave = tid >> 5;
  const float* Zb = Z + (size_t)row * 256;
  v2f zf[4], cf[4];
#pragma unroll
  for (int k4 = 0; k4 < 4; ++k4) {
    int k = k4 * 4;
    zf[k4].x = Zb[(k + hi) * 16 + lm];
    zf[k4].y = Zb[(k + 2 + hi) * 16 + lm];
    cf[k4].x = cw[lm * 16 + k + hi];        // Wt[ci][co] = conv_w[co][ci]
    cf[k4].y = cw[lm * 16 + k + 2 + hi];
  }
  float bias = cb[lm];
  lds_copy_wait();
  __syncthreads();

  for (int i = 0; i < 8; ++i) {
    int mt = wave + i * 4;                  // 32 w-tiles / 4 waves
    v8f acc = {};
#pragma unroll
    for (int k4 = 0; k4 < 4; ++k4) {        // spectral part: U x Z
      int k = k4 * 4;
      v2f a;
      a.x = Ut[(mt * 16 + lm) * 17 + k + hi * 2];
      a.y = Ut[(mt * 16 + lm) * 17 + k + hi * 2 + 1];
      acc = wmma4(a, zf[k4], acc);
    }
#pragma unroll
    for (int k4 = 0; k4 < 4; ++k4) {        // local part: h x Wt
      int k = k4 * 4;
      v2f a;
      a.x = hr[(mt * 16 + lm) * 17 + k + hi * 2];
      a.y = hr[(mt * 16 + lm) * 17 + k + hi * 2 + 1];
      acc = wmma4(a, cf[k4], acc);
    }
#pragma unroll
    for (int v = 0; v < 8; ++v)
      rb[(mt * 16 + v + 8 * hi) * 16 + lm] = acc[v] + bias;
  }
}

// ---------------------------------------------------------------------------
// Head: fc1(16->64) + bias + relu + fc2(64->1) + bias, fused.
// One 16-pixel tile per wave; 4 N-tiles of WMMA then shuffle-reduce vs fc2.
// ---------------------------------------------------------------------------
__global__ void fno_head(const float* __restrict__ hb, const float* __restrict__ w1,
                         const float* __restrict__ b1, const float* __restrict__ w2,
                         const float* __restrict__ b2, float* __restrict__ out) {
  int tid = threadIdx.x;
  int lane = tid & 31, lm = lane & 15, hi = lane >> 4, wave = tid >> 5;
  int tile = blockIdx.x * 8 + wave;
  const float* ab = hb + (size_t)tile * 16 * C_;
  v2f af[4];
#pragma unroll
  for (int k4 = 0; k4 < 4; ++k4) {
    int k = k4 * 4;
    af[k4].x = ab[lm * 16 + k + hi * 2];
    af[k4].y = ab[lm * 16 + k + hi * 2 + 1];
  }
  float sums[8];
#pragma unroll
  for (int v = 0; v < 8; ++v) sums[v] = 0.f;

#pragma unroll
  for (int nt = 0; nt < 4; ++nt) {
    v8f acc = {};
#pragma unroll
    for (int k4 = 0; k4 < 4; ++k4) {
      int k = k4 * 4;
      v2f b;
      b.x = w1[(k + hi) * HID_ + nt * 16 + lm];
      b.y = w1[(k + 2 + hi) * HID_ + nt * 16 + lm];
      acc = wmma4(af[k4], b, acc);
    }
    int j = nt * 16 + lm;
    float bj = b1[j], wj = w2[j];
#pragma unroll
    for (int v = 0; v < 8; ++v) {
      float t = fmaxf(acc[v] + bj, 0.f);
      sums[v] = fmaf(t, wj, sums[v]);
    }
  }
  // reduce over j across the 16 lanes of each half-wave
#pragma unroll
  for (int v = 0; v < 8; ++v) {
    float s = sums[v];
    s += __shfl_xor(s, 1, 16);
    s += __shfl_xor(s, 2, 16);
    s += __shfl_xor(s, 4, 16);
    s += __shfl_xor(s, 8, 16);
    sums[v] = s;
  }
  float b2v = b2[0];
#pragma unroll
  for (int v = 0; v < 8; ++v)
    if (lm == v) out[tile * 16 + 8 * hi + v] = sums[v] + b2v;
}

// ---------------------------------------------------------------------------
extern "C" void kernel_launch(void* const* d_in, const int* in_sizes, int n_in,
                              void* d_out, int out_size, void* d_ws, size_t ws_size,
                              hipStream_t stream) {
  const float* x   = (const float*)d_in[0];
  const float* w0  = (const float*)d_in[1];
  const float* b0  = (const float*)d_in[2];
  const float* swr = (const float*)d_in[3];
  const float* swi = (const float*)d_in[4];
  const float* cw  = (const float*)d_in[5];
  const float* cb  = (const float*)d_in[6];
  const float* w1  = (const float*)d_in[7];
  const float* b1  = (const float*)d_in[8];
  const float* w2  = (const float*)d_in[9];
  const float* b2  = (const float*)d_in[10];
  float* out = (float*)d_out;
  float* ws = (float*)d_ws;   // requires ~143 MB of workspace

  float* hb = ws + WS_H;
  float* G  = ws + WS_G;
  float* Z  = ws + WS_Z;
  float* TT = ws + WS_TT;
  float* UT = ws + WS_U;

  fno_init<<<64, 256, 0, stream>>>(TT, UT);
  fno_fc0<<<NPIX / 256, 256, 0, stream>>>(x, w0, b0, hb);
  for (int d = 0; d < 3; ++d) {
    fno_stageA<<<ROWS / 4, 128, 0, stream>>>(hb, TT, G);
    fno_specmix<<<B_, 256, 22528 * sizeof(float), stream>>>(
        G, swr + d * 16384, swi + d * 16384, Z);
    fno_inv2_local<<<ROWS, 128, 17408 * sizeof(float), stream>>>(
        hb, Z, UT, cw + d * 256, cb + d * 16);
  }
  fno_head<<<NPIX / 128, 256, 0, stream>>>(hb, w1, b1, w2, b2, out);
}